// TccnnV1_66228395705116
// MI455X (gfx1250) — compile-verified
//
#include <hip/hip_runtime.h>
#include <hip/hip_bf16.h>

// Problem constants (from the reference)
#define Bb 32
#define Tt 128
#define Cc 64
#define Ff 64
#define Kk 128
#define KH 3
#define KINNER 192           // KH * Ff
#define CP 62                // Cc - KH + 1
#define BN_EPS 1e-3f

// LDS layout (dynamic): weights [192][128] f32, then 2 x-tiles [66 rows][stride 66]
#define W_FLOATS   (KINNER * Kk)       // 24576
#define XT_STRIDE  66                  // 64 + 2 pad: 8B-aligned v2f, stride-2 bank walk
#define XT_SLICE   (66 * XT_STRIDE)    // rows 64,65 are read-only slop for padded M rows
#define SMEM_FLOATS (W_FLOATS + 2 * XT_SLICE)   // ~130 KB -> 2 blocks / 320KB WGP

typedef __attribute__((ext_vector_type(2))) float v2f;
typedef __attribute__((ext_vector_type(4))) float v4f;
typedef __attribute__((ext_vector_type(8))) float v8f;

extern __shared__ float smem[];

__global__ __launch_bounds__(256) void
TccnnV1_66228395705116_kernel(const float* __restrict__ x,
                              const float* __restrict__ conv_w,
                              const float* __restrict__ conv_b,
                              const float* __restrict__ bn_gamma,
                              const float* __restrict__ bn_beta,
                              const float* __restrict__ bn_mean,
                              const float* __restrict__ bn_var,
                              float* __restrict__ out) {
    float* bw = smem;              // [KINNER][Kk] row-major (same layout as conv_w)
    float* xt = smem + W_FLOATS;   // 2 slices of [64+2][XT_STRIDE]

    const int tid = threadIdx.x;
    const int blk = blockIdx.x;                  // 0 .. B*T/2 - 1 (two t-slices per block)

    // ---- Stage weights into LDS: 24576 floats = 6144 v4f, 24 per thread ----
    {
        const v4f* wsrc = (const v4f*)conv_w;
        v4f* wdst = (v4f*)bw;
        #pragma unroll
        for (int i = 0; i < 24; ++i)
            wdst[tid + i * 256] = wsrc[tid + i * 256];
    }

    // ---- Stage two x tiles (2 x 64x64) into LDS with row stride 66 ----
    {
        const v2f* xsrc = (const v2f*)(x + (size_t)blk * (2 * Cc * Ff));
        #pragma unroll
        for (int i = 0; i < 16; ++i) {
            int idx = tid + i * 256;          // 0..4095 v2f elements (2 slices)
            int s   = idx >> 11;              // slice 0/1
            int rem = idx & 2047;
            int c   = rem >> 5;               // row 0..63
            int j   = rem & 31;               // v2f col
            *(v2f*)&xt[s * XT_SLICE + c * XT_STRIDE + 2 * j] = xsrc[idx];
        }
    }
    __syncthreads();

    // ---- Wave tiling: 8 waves, each owns 2 M-tiles x 4 N-tiles (of 64 tiles) ----
    const int wave = tid >> 5;
    const int lane = tid & 31;
    const int l15  = lane & 15;
    const int hi   = lane >> 4;          // 0: K lo pair, 1: K hi pair
    const int q    = wave & 3;
    const int sl   = q >> 1;             // which t-slice (0/1)
    const int mh   = q & 1;              // which 32-row band inside the slice
    const int nh   = wave >> 2;          // N half (0 or 1)
    const float* xts = xt + sl * XT_SLICE;
    const int mA0  = 32 * mh + l15;      // A row of lane for M-tile 0 (tile 1 = +16)

    v8f acc[2][4] = {};

    #pragma unroll 2
    for (int kk = 0; kk < KINNER / 4; ++kk) {
        const int kidx = 4 * kk + 2 * hi;       // K index of this lane's A pair
        const int kh   = kidx >> 6;             // which x row offset (0..2)
        const int kw   = kidx & 63;             // even -> v2f 8B aligned
        // A frags: x_tile rows (mA0+kh), (mA0+16+kh), cols kw..kw+1 (implicit im2col)
        v2f a0 = *(const v2f*)&xts[(mA0 + kh) * XT_STRIDE + kw];
        v2f a1 = *(const v2f*)&xts[(mA0 + 16 + kh) * XT_STRIDE + kw];

        // B frags: row kidx(+1), col = 64*nh + 16n + l15; rows stride 128 floats
        const float* bp = &bw[kidx * Kk + 64 * nh + l15];
        v2f b0; b0.x = bp[0];  b0.y = bp[Kk];
        v2f b1; b1.x = bp[16]; b1.y = bp[16 + Kk];
        v2f b2; b2.x = bp[32]; b2.y = bp[32 + Kk];
        v2f b3; b3.x = bp[48]; b3.y = bp[48 + Kk];

        acc[0][0] = __builtin_amdgcn_wmma_f32_16x16x4_f32(false, a0, false, b0, (short)0, acc[0][0], false, false);
        acc[0][1] = __builtin_amdgcn_wmma_f32_16x16x4_f32(false, a0, false, b1, (short)0, acc[0][1], false, false);
        acc[0][2] = __builtin_amdgcn_wmma_f32_16x16x4_f32(false, a0, false, b2, (short)0, acc[0][2], false, false);
        acc[0][3] = __builtin_amdgcn_wmma_f32_16x16x4_f32(false, a0, false, b3, (short)0, acc[0][3], false, false);
        acc[1][0] = __builtin_amdgcn_wmma_f32_16x16x4_f32(false, a1, false, b0, (short)0, acc[1][0], false, false);
        acc[1][1] = __builtin_amdgcn_wmma_f32_16x16x4_f32(false, a1, false, b1, (short)0, acc[1][1], false, false);
        acc[1][2] = __builtin_amdgcn_wmma_f32_16x16x4_f32(false, a1, false, b2, (short)0, acc[1][2], false, false);
        acc[1][3] = __builtin_amdgcn_wmma_f32_16x16x4_f32(false, a1, false, b3, (short)0, acc[1][3], false, false);
    }

    // ---- Epilogue: bias + BN(t) + ReLU, masked store (rows < 62 per slice) ----
    const int tslice = ((blk << 1) + sl) & (Tt - 1);   // BN parameter index
    const float sc = bn_gamma[tslice] * rsqrtf(bn_var[tslice] + BN_EPS);
    const float sh = bn_beta[tslice] - bn_mean[tslice] * sc;

    float* obase = out + (size_t)((blk << 1) + sl) * CP * Kk;   // out[b][t][c'][k]
    const int rbase = 32 * mh + 8 * hi;                          // + 16*mi + v

    #pragma unroll
    for (int mi = 0; mi < 2; ++mi) {
        #pragma unroll
        for (int n = 0; n < 4; ++n) {
            const int col  = 16 * (4 * nh + n) + l15;   // k index
            const float bias = conv_b[col];
            #pragma unroll
            for (int v = 0; v < 8; ++v) {
                const int row = rbase + 16 * mi + v;    // c' index within slice
                if (row < CP) {
                    float y = (acc[mi][n][v] + bias) * sc + sh;
                    y = fmaxf(y, 0.0f);
                    obase[(size_t)row * Kk + col] = y;
                }
            }
        }
    }
}

extern "C" void kernel_launch(void* const* d_in, const int* in_sizes, int n_in,
                              void* d_out, int out_size, void* d_ws, size_t ws_size,
                              hipStream_t stream) {
    const float* x        = (const float*)d_in[0];
    const float* conv_w   = (const float*)d_in[1];
    const float* conv_b   = (const float*)d_in[2];
    const float* bn_gamma = (const float*)d_in[3];
    const float* bn_beta  = (const float*)d_in[4];
    const float* bn_mean  = (const float*)d_in[5];
    const float* bn_var   = (const float*)d_in[6];
    float* out = (float*)d_out;

    const size_t smem_bytes = SMEM_FLOATS * sizeof(float);   // ~130 KB dynamic LDS
    TccnnV1_66228395705116_kernel<<<dim3(Bb * Tt / 2), dim3(256), smem_bytes, stream>>>(
        x, conv_w, conv_b, bn_gamma, bn_beta, bn_mean, bn_var, out);
}